// CenterPointTarget_83021717832052
// MI455X (gfx1250) — compile-verified
//
#include <hip/hip_runtime.h>
#include <cstdint>

#define Bz 8
#define Nn 128
#define Cc 2
#define Hh 512
#define Ww 512
#define HW (Hh*Ww)
#define BD 10
#define R_MAX 8
#define WIN 17
#define OSF 4.0f
#define LDS_STRIDE 528   // 512 + pad to break bank conflicts

__device__ __forceinline__ int iabs_(int v) { return v < 0 ? -v : v; }

__device__ __forceinline__ float gaussian_radius_dev(float h, float w) {
    const float mo = 0.1f;
    float b1 = h + w;
    float c1 = w * h * (1.0f - mo) / (1.0f + mo);
    float r1 = (b1 + sqrtf(fmaxf(b1 * b1 - 4.0f * c1, 0.0f))) * 0.5f;
    float b2 = 2.0f * (h + w);
    float c2 = (1.0f - mo) * w * h;
    float r2 = (b2 + sqrtf(fmaxf(b2 * b2 - 16.0f * c2, 0.0f))) * 0.5f;
    float a3 = 4.0f * mo;
    float b3 = -2.0f * mo * (h + w);
    float c3 = (mo - 1.0f) * w * h;
    float r3 = (b3 + sqrtf(fmaxf(b3 * b3 - 4.0f * a3 * c3, 0.0f))) * 0.5f;
    return fminf(r1, fminf(r2, r3));
}

// ---------------- zero fill ----------------
__global__ void zero_kernel(float* __restrict__ p, size_t n) {
    size_t i = (size_t)blockIdx.x * blockDim.x + threadIdx.x;
    size_t st = (size_t)gridDim.x * blockDim.x;
    for (; i < n; i += st) p[i] = 0.0f;
}

// ---------------- logits = clip(sigmoid(x), eps, 1-eps) ----------------
__global__ void logits_kernel(const float* __restrict__ x, float* __restrict__ o, size_t n) {
    size_t i = (size_t)blockIdx.x * blockDim.x + threadIdx.x;
    size_t st = (size_t)gridDim.x * blockDim.x;
    for (; i < n; i += st) {
        float s = 1.0f / (1.0f + expf(-x[i]));
        o[i] = fminf(fmaxf(s, 1e-4f), 1.0f - 1e-4f);
    }
}

// ---------------- bbox_pred: NCHW channels -> (B,H,W,10) via async-to-LDS staging ----------------
__global__ __launch_bounds__(512) void bboxpred_kernel(const float* __restrict__ reg,
                                                       const float* __restrict__ hei,
                                                       const float* __restrict__ dim,
                                                       const float* __restrict__ rot,
                                                       const float* __restrict__ vel,
                                                       float* __restrict__ out) {
    __shared__ float sm[BD * LDS_STRIDE];
    int row = blockIdx.x;                 // b*H + h
    int b = row / Hh, h = row % Hh;
    int t = threadIdx.x;                  // 0..511  (one per pixel in the row)

    const float* chan[BD];
    chan[0] = reg + (size_t)b * 2 * HW;
    chan[1] = reg + (size_t)b * 2 * HW + HW;
    chan[2] = hei + (size_t)b * 1 * HW;
    chan[3] = dim + (size_t)b * 3 * HW;
    chan[4] = dim + (size_t)b * 3 * HW + HW;
    chan[5] = dim + (size_t)b * 3 * HW + 2 * HW;
    chan[6] = rot + (size_t)b * 2 * HW;
    chan[7] = rot + (size_t)b * 2 * HW + HW;
    chan[8] = vel + (size_t)b * 2 * HW;
    chan[9] = vel + (size_t)b * 2 * HW + HW;

    // Async copy 10 channel segments (512 floats each) into LDS.
    #pragma unroll
    for (int c = 0; c < BD; ++c) {
        unsigned lds_off = (unsigned)(unsigned long long)(uintptr_t)(&sm[c * LDS_STRIDE + t]);
        unsigned long long ga = (unsigned long long)(uintptr_t)(chan[c] + (size_t)h * Ww + t);
        asm volatile("global_load_async_to_lds_b32 %0, %1, off"
                     :: "v"(lds_off), "v"(ga) : "memory");
    }
    asm volatile("s_wait_asynccnt 0" ::: "memory");
    __syncthreads();

    // Fully-coalesced interleaved write-out: element (p, c) at base + p*10 + c.
    size_t base = (size_t)row * (Ww * BD);
    #pragma unroll
    for (int k = 0; k < BD; ++k) {
        int oi = k * 512 + t;
        int c = oi % BD;
        int p = oi / BD;
        out[base + oi] = sm[c * LDS_STRIDE + p];
    }
}

// ---------------- heat: per-object gaussian max-scatter (order independent) ----------------
__global__ __launch_bounds__(320) void heat_kernel(const float* __restrict__ gt, float* __restrict__ heat) {
    int obj = blockIdx.x;                 // b*N + n
    int b = obj / Nn, n = obj % Nn;
    const float* g = gt + (size_t)b * Nn * 10 + (size_t)n * 10;
    float cx = g[0] / OSF, cy = g[1] / OSF;
    float fw = g[3] / OSF, fl = g[4] / OSF;
    int xi = (int)cx, yi = (int)cy;       // truncation matches astype(int32) for positives
    int rad = (int)gaussian_radius_dev(fl, fw);
    if (rad < 2) rad = 2;
    if (rad > R_MAX) rad = R_MAX;
    bool valid = (fw > 0.0f) && (fl > 0.0f) && (xi >= 0) && (xi < Ww) && (yi >= 0) && (yi < Hh);
    int cls = (int)g[9];
    cls = cls < 0 ? 0 : (cls > Cc - 1 ? Cc - 1 : cls);
    float sigma = (float)(2 * rad + 1) / 6.0f;
    float inv2s2 = 1.0f / (2.0f * sigma * sigma);

    int t = threadIdx.x;
    if (t >= WIN * WIN || !valid) return;
    int oi = t / WIN - R_MAX;             // row offset (a)
    int oj = t % WIN - R_MAX;             // col offset (b)
    if (iabs_(oi) > rad || iabs_(oj) > rad) return;
    int rr = yi + oi, cc = xi + oj;
    if (rr < 0 || rr >= Hh || cc < 0 || cc >= Ww) return;
    float gv = expf(-(float)(oi * oi + oj * oj) * inv2s2);
    // heat >= 0, so int-bitpattern atomicMax == float max
    atomicMax((int*)(heat + (size_t)b * Cc * HW + (size_t)cls * HW + (size_t)rr * Ww + cc),
              __float_as_int(gv));
}

// ---------------- bbox_targets: sequential last-writer-wins scatter in stable class order ----------------
__global__ __launch_bounds__(320) void targets_kernel(const float* __restrict__ gt, float* __restrict__ bt) {
    int b = blockIdx.x;
    int t = threadIdx.x;
    for (int key = 0; key < Cc; ++key) {          // stable sort by class == class-major, index-minor
        for (int n = 0; n < Nn; ++n) {
            const float* g = gt + (size_t)b * Nn * 10 + (size_t)n * 10;
            int cls = (int)g[9];
            if (cls != key) continue;             // uniform across block
            float cx = g[0] / OSF, cy = g[1] / OSF;
            float fw = g[3] / OSF, fl = g[4] / OSF;
            int xi = (int)cx, yi = (int)cy;
            int rad = (int)gaussian_radius_dev(fl, fw);
            if (rad < 2) rad = 2;
            if (rad > R_MAX) rad = R_MAX;
            bool valid = (fw > 0.0f) && (fl > 0.0f) && (xi >= 0) && (xi < Ww) && (yi >= 0) && (yi < Hh);
            if (t < WIN * WIN) {
                int oi = t / WIN - R_MAX;
                int oj = t % WIN - R_MAX;
                int rr = yi + oi, cc = xi + oj;
                bool ok = valid && iabs_(oi) <= rad && iabs_(oj) <= rad &&
                          rr >= 0 && rr < Hh && cc >= 0 && cc < Ww;
                if (ok) {
                    // replicate reference exactly (ch0 uses row offset with x, ch1 col offset with y)
                    float v0 = cx - (float)(xi + oi);
                    float v1 = cy - (float)(yi + oj);
                    float lw = logf(fmaxf(g[3], 1e-6f));
                    float ll = logf(fmaxf(g[4], 1e-6f));
                    float lh = logf(fmaxf(g[5], 1e-6f));
                    float sr = sinf(g[6]), cr = cosf(g[6]);
                    float* o = bt + ((size_t)b * HW + (size_t)rr * Ww + cc) * BD;
                    o[0] = v0; o[1] = v1; o[2] = g[2]; o[3] = lw; o[4] = ll;
                    o[5] = lh; o[6] = sr; o[7] = cr; o[8] = g[7]; o[9] = g[8];
                }
            }
            __threadfence();     // order this object's stores before the next object's (WAW)
            __syncthreads();
        }
    }
}

// ---------------- bbox_weight + deterministic per-block partial sums ----------------
__global__ __launch_bounds__(256) void weight_kernel(const float* __restrict__ heat,
                                                     float* __restrict__ bw,
                                                     float* __restrict__ partial) {
    __shared__ float sred[256];
    int tid = threadIdx.x;
    size_t gidx = (size_t)blockIdx.x * blockDim.x + tid;
    size_t st = (size_t)gridDim.x * blockDim.x;
    float local = 0.0f;
    for (size_t i = gidx; i < (size_t)Bz * HW; i += st) {
        size_t b = i / HW, hw = i % HW;
        float m = fmaxf(heat[b * Cc * HW + hw], heat[b * Cc * HW + HW + hw]);
        float* o = bw + i * BD;
        #pragma unroll
        for (int c = 0; c < BD; ++c) o[c] = m;
        local += m;
    }
    sred[tid] = local;
    __syncthreads();
    for (int s = 128; s > 0; s >>= 1) {
        if (tid < s) sred[tid] += sred[tid + s];
        __syncthreads();
    }
    if (tid == 0) partial[blockIdx.x] = sred[0];
}

// ---------------- final reduction of 1024 partials via V_WMMA_F32_16X16X4_F32 ----------------
// A = ones(16x4) in full f32, so D[m][j] = sum_k B[k][j] (+C): exact sum regardless of
// the lane->(k,j) mapping. C accumulates across iterations; wave-sum of c[0] = 2*total.
typedef __attribute__((ext_vector_type(2))) float v2f;
typedef __attribute__((ext_vector_type(8))) float v8f;

__global__ __launch_bounds__(32) void finalize_kernel(const float* __restrict__ partial,
                                                      float* __restrict__ avg) {
    int lane = threadIdx.x;               // one full wave32, EXEC all ones
    v2f a; a.x = 1.0f; a.y = 1.0f;
    v8f c = {};
    #pragma unroll
    for (int it = 0; it < 16; ++it) {     // 16 iters * 64 values = 1024 partials
        v2f bb;
        bb.x = partial[it * 64 + lane];
        bb.y = partial[it * 64 + 32 + lane];
        c = __builtin_amdgcn_wmma_f32_16x16x4_f32(false, a, false, bb, (short)0, c, false, false);
    }
    float s = c[0];
    #pragma unroll
    for (int m = 16; m >= 1; m >>= 1) s += __shfl_xor(s, m, 32);
    s *= 0.5f;                            // rows 0 and 8 both present in c[0] -> halve (exact)
    if (lane == 0) avg[0] = fmaxf(s, 1.0f);
}

extern "C" void kernel_launch(void* const* d_in, const int* in_sizes, int n_in,
                              void* d_out, int out_size, void* d_ws, size_t ws_size,
                              hipStream_t stream) {
    (void)in_sizes; (void)n_in; (void)out_size; (void)ws_size;
    const float* gt  = (const float*)d_in[0];
    const float* hm  = (const float*)d_in[1];
    const float* reg = (const float*)d_in[2];
    const float* hei = (const float*)d_in[3];
    const float* dim = (const float*)d_in[4];
    const float* rot = (const float*)d_in[5];
    const float* vel = (const float*)d_in[6];

    float* out      = (float*)d_out;
    float* o_logits = out;                                   // (B,C,H,W)
    float* o_heat   = o_logits + (size_t)Bz * Cc * HW;       // (B,C,H,W)
    float* o_bpred  = o_heat   + (size_t)Bz * Cc * HW;       // (B,H,W,10)
    float* o_btgt   = o_bpred  + (size_t)Bz * HW * BD;       // (B,H,W,10)
    float* o_bwt    = o_btgt   + (size_t)Bz * HW * BD;       // (B,H,W,10)
    float* o_avg    = o_bwt    + (size_t)Bz * HW * BD;       // scalar
    float* ws_partial = (float*)d_ws;                        // 1024 floats

    zero_kernel<<<2048, 256, 0, stream>>>(o_heat, (size_t)Bz * Cc * HW);
    zero_kernel<<<4096, 256, 0, stream>>>(o_btgt, (size_t)Bz * HW * BD);
    logits_kernel<<<4096, 256, 0, stream>>>(hm, o_logits, (size_t)Bz * Cc * HW);
    bboxpred_kernel<<<Bz * Hh, 512, 0, stream>>>(reg, hei, dim, rot, vel, o_bpred);
    heat_kernel<<<Bz * Nn, 320, 0, stream>>>(gt, o_heat);
    targets_kernel<<<Bz, 320, 0, stream>>>(gt, o_btgt);
    weight_kernel<<<1024, 256, 0, stream>>>(o_heat, o_bwt, ws_partial);
    finalize_kernel<<<1, 32, 0, stream>>>(ws_partial, o_avg);
}